// Peaks3d_25829933318397
// MI455X (gfx1250) — compile-verified
//
#include <hip/hip_runtime.h>
#include <cmath>

// Problem geometry (fixed by the reference): x is (B=4, C=4, D=24, H=384, W=384) fp32.
#define B_N     4
#define C_N     4
#define D_N     24
#define HWN     147456              // 384*384, h/w are jointly contiguous per (b,c,d)
#define TW      256                 // hw positions per block tile
#define NSLICE  (C_N * D_N)         // 96 (c,d) slices per tile
#define TILES_PER_B (HWN / TW)      // 576

struct W7 { float w[7]; };

// ---------------------------------------------------------------------------
// Bandwidth-bound fused kernel:
//   stage (4c x 24d x 256w) = 96 KB tile to LDS via async DMA loads (ASYNCcnt),
//   then channel-sum + 7-tap depth conv + ReLU + broadcast-store to 4 channels.
// ---------------------------------------------------------------------------
__global__ __launch_bounds__(256) void peaks3d_kernel(const float* __restrict__ x,
                                                      float* __restrict__ out,
                                                      W7 wt) {
    __shared__ float lds[NSLICE * TW];          // 96 KB: lds[(c*24+d)*256 + w]

    const int tid = threadIdx.x;
    const int bid = blockIdx.x;
    const int b   = bid / TILES_PER_B;
    const int hw0 = (bid % TILES_PER_B) * TW;

    // Low 32 bits of the generic shared-pointer = LDS byte offset (aperture scheme).
    const unsigned lbase = (unsigned)(unsigned long long)(&lds[0]);
    const unsigned long long xbase = (unsigned long long)(const char*)x;

    // ---- async global -> LDS copy of the whole 96 KB tile ----
    // flat b128 index i: slice = i>>6 (16B*64 = one 1KB slice row), w16 = i&63.
    // Per wave-instruction: 32 lanes x 16 B = 512 contiguous bytes.
#pragma unroll
    for (int j = 0; j < 24; ++j) {
        const int i     = j * 256 + tid;
        const int slice = i >> 6;               // c*24 + d
        const int w16   = i & 63;
        const unsigned lofs = lbase + (unsigned)i * 16u;
        const unsigned long long gaddr =
            xbase + ((unsigned long long)(b * NSLICE + slice) * HWN
                     + (unsigned)(hw0 + w16 * 4)) * 4ull;
        asm volatile("global_load_async_to_lds_b128 %0, %1, off"
                     :: "v"(lofs), "v"(gaddr)
                     : "memory");
    }
    // Wait for this wave's async DMA writes to land in LDS, then sync the block.
    asm volatile("s_wait_asynccnt 0x0" ::: "memory");
    __syncthreads();

    // ---- channel sum into registers (bank-conflict-free: stride 1 across lanes) ----
    float s[D_N];
#pragma unroll
    for (int d = 0; d < D_N; ++d) {
        s[d] = lds[(0 * D_N + d) * TW + tid]
             + lds[(1 * D_N + d) * TW + tid]
             + lds[(2 * D_N + d) * TW + tid]
             + lds[(3 * D_N + d) * TW + tid];
    }

    // ---- 7-tap zero-padded depth conv + ReLU + broadcast to 4 channels ----
    const unsigned long long obase =
        (unsigned long long)(b * NSLICE) * HWN + (unsigned)(hw0 + tid);
    const unsigned long long cstride = (unsigned long long)D_N * HWN;

#pragma unroll
    for (int d = 0; d < D_N; ++d) {
        float acc = 0.0f;
#pragma unroll
        for (int k = 0; k < 7; ++k) {
            const int dd = d + k - 3;
            if (dd >= 0 && dd < D_N) acc = fmaf(wt.w[k], s[dd], acc);
        }
        acc = fmaxf(acc, 0.0f);
        float* o = out + obase + (unsigned long long)d * HWN;
        o[0]           = acc;
        o[cstride]     = acc;
        o[2 * cstride] = acc;
        o[3 * cstride] = acc;
    }
}

// Host-side replication of the reference fp32 weight construction.
static void make_weights(float w[7]) {
    const float SIGMA = 1.5f;
    float xs[7];
    for (int i = 0; i < 7; ++i) xs[i] = -9.0f + (19.0f / 6.0f) * (float)i;  // linspace(-9, 10, 7)
    float mean = 0.0f;
    for (int i = 0; i < 7; ++i) {
        const float x2 = xs[i] * xs[i];
        const float g  = 1.0f / (sqrtf(2.0f * (float)M_PI) * SIGMA)
                         * expf(-x2 / (2.0f * SIGMA * SIGMA));
        w[i] = g * (x2 / (SIGMA * SIGMA * SIGMA * SIGMA) - 1.0f) / (SIGMA * SIGMA);
        mean += w[i];
    }
    mean /= 7.0f;
    float norm = 0.0f;
    for (int i = 0; i < 7; ++i) {
        w[i] -= mean;
        norm += w[i] * xs[i] * xs[i];
    }
    norm *= 0.5f;
    for (int i = 0; i < 7; ++i) w[i] = -(w[i] / norm);
}

extern "C" void kernel_launch(void* const* d_in, const int* in_sizes, int n_in,
                              void* d_out, int out_size, void* d_ws, size_t ws_size,
                              hipStream_t stream) {
    (void)in_sizes; (void)n_in; (void)d_ws; (void)ws_size; (void)out_size;

    const float* x = (const float*)d_in[0];
    float* out     = (float*)d_out;

    W7 wt;
    make_weights(wt.w);

    const dim3 grid(B_N * TILES_PER_B);   // 4 * 576 = 2304 blocks
    const dim3 block(256);
    hipLaunchKernelGGL(peaks3d_kernel, grid, block, 0, stream, x, out, wt);
}